// KittyCatConv_44564580663867
// MI455X (gfx1250) — compile-verified
//
#include <hip/hip_runtime.h>
#include <math.h>

// ---------------- workspace layout (float offsets) ----------------
#define OFF_PROJ 0u        // [2][8][8][1024]            = 131072
#define OFF_CONV 131072u   // [2][4][8][8][1024]         = 524288
#define OFF_BNP  655360u   // [2][4][8][2] scale,shift   = 128
#define OFF_QT   655488u   // [64][1024]                 = 65536
#define OFF_KT   721024u   // [64][1024]                 = 65536
#define OFF_C    786560u   // cscale scalar

typedef float v2f __attribute__((ext_vector_type(2)));
typedef float v8f __attribute__((ext_vector_type(8)));

// ---------------- 1) projection: q = Q . W_pq, k = K . W_pq ----------------
__global__ void k_proj(const float* __restrict__ Q, const float* __restrict__ K,
                       const float* __restrict__ Wpq, float* __restrict__ ws) {
  int gid = blockIdx.x * 256 + threadIdx.x;   // 131072 total
  int stream = gid >> 16;
  int rem = gid & 65535;                      // bh*1024 + l
  const float* src = (stream == 0) ? Q : K;
  const float4* v = (const float4*)(src + (size_t)rem * 64);
  const float4* w4 = (const float4*)Wpq;
  float acc = 0.f;
#pragma unroll
  for (int j = 0; j < 16; ++j) {
    float4 a = v[j], w = w4[j];
    acc += a.x * w.x + a.y * w.y + a.z * w.z + a.w * w.w;
  }
  ws[OFF_PROJ + stream * 65536 + rem] = acc;
}

// ---------------- 2) conv1d, 'same' padding, one (stream,filter); grid=8 (batch) ----
__global__ void k_conv(const float* __restrict__ x, const float* __restrict__ w,
                       const float* __restrict__ bias, float* __restrict__ y, int F) {
  __shared__ float xs[8][1024];
  __shared__ float wl[8 * 8 * 9];
  __shared__ float bl[8];
  int b = blockIdx.x, tid = threadIdx.x;
  const float* xb = x + b * 8192;             // [cin][t]
  for (int p = tid; p < 8192; p += 256) xs[p >> 10][p & 1023] = xb[p];
  for (int p = tid; p < 64 * F; p += 256) wl[p] = w[p];
  if (tid < 8) bl[tid] = bias[tid];
  __syncthreads();
  int pad = (F - 1) >> 1;
  for (int it = 0; it < 32; ++it) {
    int o = it * 256 + tid;
    int c = o >> 10, t = o & 1023;
    float acc = bl[c];
    for (int cin = 0; cin < 8; ++cin)
      for (int j = 0; j < F; ++j) {
        int tt = t + j - pad;
        if (tt >= 0 && tt < 1024) acc += wl[(c * 8 + cin) * F + j] * xs[cin][tt];
      }
    y[(b * 8 + c) * 1024 + t] = acc;
  }
}

// ---------------- 3) BN training stats -> scale/shift; grid=64 = s*32+i*8+c ------
__global__ void k_bnstats(float* __restrict__ ws, const float* __restrict__ g,
                          const float* __restrict__ beta) {
  int grp = blockIdx.x;
  int c = grp & 7;
  const float* y = ws + OFF_CONV + (grp >> 3) * 65536;   // (s*4+i) block, [b][c][t]
  int tid = threadIdx.x;
  float s = 0.f, s2 = 0.f;
  for (int it = 0; it < 32; ++it) {
    int idx = it * 256 + tid;                 // b*1024 + t
    int b = idx >> 10, t = idx & 1023;
    float v = y[(b * 8 + c) * 1024 + t];
    s += v; s2 += v * v;
  }
  __shared__ float sa[256], sb[256];
  sa[tid] = s; sb[tid] = s2; __syncthreads();
  for (int st = 128; st > 0; st >>= 1) {
    if (tid < st) { sa[tid] += sa[tid + st]; sb[tid] += sb[tid + st]; }
    __syncthreads();
  }
  if (tid == 0) {
    float m = sa[0] * (1.f / 8192.f);
    float var = sb[0] * (1.f / 8192.f) - m * m;
    float rstd = rsqrtf(var + 1e-5f);
    float sc = g[c] * rstd;
    ws[OFF_BNP + grp * 2] = sc;
    ws[OFF_BNP + grp * 2 + 1] = beta[c] - m * sc;
  }
}

// ---------------- 4) cscale = dot(W_bq, W_bk)/sqrt(dk) ----------------
__global__ void k_cscale(const float* __restrict__ wbq, const float* __restrict__ wbk,
                         float* __restrict__ ws) {
  __shared__ float sm[64];
  int l = threadIdx.x;
  sm[l] = wbq[l] * wbk[l];
  __syncthreads();
  if (l < 32) sm[l] += sm[l + 32];
  __syncthreads();
  if (l == 0) {
    float t = 0.f;
    for (int i = 0; i < 32; ++i) t += sm[i];
    ws[OFF_C] = t * 0.125f;
  }
}

// ---------------- bitonic sort (descending) on shared array ----------------
__device__ inline void bitonic_desc(float* a, int n) {
  for (int k = 2; k <= n; k <<= 1)
    for (int j = k >> 1; j > 0; j >>= 1) {
      __syncthreads();
      for (int i = threadIdx.x; i < n; i += blockDim.x) {
        int ixj = i ^ j;
        if (ixj > i) {
          float x = a[i], yv = a[ixj];
          bool seg = (i & k) != 0;
          if (seg ? (x > yv) : (x < yv)) { a[i] = yv; a[ixj] = x; }
        }
      }
    }
  __syncthreads();
}

// ---------------- 5a) Q: gather(flat-reorder)+BN+ELU, sort 4096 desc, keep 1024 ----
__global__ void k_sortQ(float* __restrict__ ws) {
  __shared__ float a[4096];
  int bh = blockIdx.x, b2 = bh >> 3, h2 = bh & 7;
  int A = b2 * 4 + (h2 >> 1);
  int fi = A >> 3, bb = A & 7;
  const float* y = ws + OFF_CONV + fi * 65536;           // stream 0
  int tid = threadIdx.x;
  for (int p = tid; p < 4096; p += 256) {
    int j = p >> 10, t = p & 1023;
    int c = (h2 & 1) * 4 + j;
    float sc = ws[OFF_BNP + (fi * 8 + c) * 2];
    float sh = ws[OFF_BNP + (fi * 8 + c) * 2 + 1];
    float v = y[(bb * 8 + c) * 1024 + t] * sc + sh;
    a[p] = v > 0.f ? v : expm1f(v);
  }
  bitonic_desc(a, 4096);
  for (int p = tid; p < 1024; p += 256) ws[OFF_QT + bh * 1024 + p] = a[p];
}

// ---------------- 5b) K: gather, filter-mean, sort 1024 desc ----------------
__global__ void k_sortK(float* __restrict__ ws) {
  __shared__ float a[1024];
  int bh = blockIdx.x, b2 = bh >> 3, h2 = bh & 7;
  int A = b2 * 4 + (h2 >> 1);
  int fi = A >> 3, bb = A & 7;
  const float* y = ws + OFF_CONV + (4 + fi) * 65536;     // stream 1
  int tid = threadIdx.x;
  for (int t = tid; t < 1024; t += 256) {
    float acc = 0.f;
    for (int j = 0; j < 4; ++j) {
      int c = (h2 & 1) * 4 + j;
      float sc = ws[OFF_BNP + (32 + fi * 8 + c) * 2];
      float sh = ws[OFF_BNP + (32 + fi * 8 + c) * 2 + 1];
      float v = y[(bb * 8 + c) * 1024 + t] * sc + sh;
      acc += (v > 0.f ? v : expm1f(v));
    }
    a[t] = acc * 0.25f;
  }
  bitonic_desc(a, 1024);
  for (int t = tid; t < 1024; t += 256) ws[OFF_KT + bh * 1024 + t] = a[t];
}

// ---------------- 6) heavy: softmax of rank-1 scores -> attn + WMMA context ------
#define KCH 64
__global__ void k_attn(const float* __restrict__ ws, const float* __restrict__ V,
                       float* __restrict__ ctx, float* __restrict__ attn) {
  __shared__ float sKt[1024];
  __shared__ float sVt[64][KCH + 2];          // V transposed: [n][k], padded
  __shared__ float sE[8][16][KCH + 2];        // per-wave exp tile
  __shared__ float sP[8][16][3];              // s, mx, invden per row
  int bh = blockIdx.x >> 3;
  int q0 = (blockIdx.x & 7) * 128;
  int tid = threadIdx.x, wave = tid >> 5, lane = tid & 31;
  int half = lane >> 4, r = lane & 15;
  const float* Kt = ws + OFF_KT + bh * 1024;
  const float* Qt = ws + OFF_QT + bh * 1024;
  float csc = ws[OFF_C];
  for (int p = tid; p < 1024; p += 256) sKt[p] = Kt[p];
  __syncthreads();

  int row = q0 + wave * 16 + r;
  float s = csc * Qt[row];
  float mx = (s >= 0.f) ? s * sKt[0] : s * sKt[1023];   // Kt sorted descending
  float part = 0.f;
  for (int k = half; k < 1024; k += 2) part += __expf(s * sKt[k] - mx);
  part += __shfl_xor(part, 16, 32);
  float invden = 1.f / part;
  if (half == 0) { sP[wave][r][0] = s; sP[wave][r][1] = mx; sP[wave][r][2] = invden; }

  const float* Vb = V + (size_t)bh * 65536;
  float* attnB = attn + (size_t)bh * 1024 * 1024;
  v8f acc[4];
#pragma unroll
  for (int nt = 0; nt < 4; ++nt) acc[nt] = (v8f)(0.0f);

  for (int kb = 0; kb < 1024; kb += KCH) {
    __syncthreads();
    // stage V chunk transposed into LDS: 64k x 64n
    for (int it = 0; it < 4; ++it) {
      int idx = tid + 256 * it;               // 1024 float4s
      int k = idx >> 4, n4 = (idx & 15) * 4;
      float4 vv = *(const float4*)(Vb + (size_t)(kb + k) * 64 + n4);
      sVt[n4 + 0][k] = vv.x; sVt[n4 + 1][k] = vv.y;
      sVt[n4 + 2][k] = vv.z; sVt[n4 + 3][k] = vv.w;
    }
    __syncthreads();
    // compute normalized exp tile once; write attn (coalesced, nontemporal) + LDS
    for (int rr = 0; rr < 16; ++rr) {
      float sw = sP[wave][rr][0], mw = sP[wave][rr][1], iw = sP[wave][rr][2];
      int rowr = q0 + wave * 16 + rr;
#pragma unroll
      for (int j = 0; j < KCH / 32; ++j) {
        int kl = j * 32 + lane;
        float e = __expf(sw * sKt[kb + kl] - mw) * iw;
        sE[wave][rr][kl] = e;
        __builtin_nontemporal_store(e, &attnB[(size_t)rowr * 1024 + kb + kl]);
      }
    }
    // E(16 x KCH) @ V(KCH x 64) via v_wmma_f32_16x16x4_f32
    for (int k0 = 0; k0 < KCH; k0 += 4) {
      int ka = k0 + 2 * half;                 // lanes 0-15: K,K+1 ; 16-31: K+2,K+3
      v2f a = *(const v2f*)&sE[wave][r][ka];
#pragma unroll
      for (int nt = 0; nt < 4; ++nt) {
        v2f bfrag = *(const v2f*)&sVt[nt * 16 + r][ka];
        acc[nt] = __builtin_amdgcn_wmma_f32_16x16x4_f32(
            false, a, false, bfrag, (short)0, acc[nt], false, false);
      }
    }
  }
  // context epilogue: C/D layout lanes0-15 M=g, lanes16-31 M=g+8, N=lane&15
  float* ctxB = ctx + (size_t)bh * 65536;
#pragma unroll
  for (int nt = 0; nt < 4; ++nt) {
    int n = nt * 16 + r;
#pragma unroll
    for (int m = 0; m < 8; ++m) {
      int rowm = q0 + wave * 16 + m + 8 * half;
      ctxB[(size_t)rowm * 64 + n] = acc[nt][m];
    }
  }
}

// ---------------- launch ----------------
extern "C" void kernel_launch(void* const* d_in, const int* in_sizes, int n_in,
                              void* d_out, int out_size, void* d_ws, size_t ws_size,
                              hipStream_t stream) {
  (void)in_sizes; (void)n_in; (void)out_size; (void)ws_size;
  const float* Q    = (const float*)d_in[0];
  const float* K    = (const float*)d_in[1];
  const float* V    = (const float*)d_in[2];
  const float* Wpq  = (const float*)d_in[3];
  const float* Wbq  = (const float*)d_in[4];
  const float* Wbk  = (const float*)d_in[5];
  const float* bn_g = (const float*)d_in[6];
  const float* bn_b = (const float*)d_in[7];
  float* ws  = (float*)d_ws;
  float* out = (float*)d_out;

  k_proj<<<512, 256, 0, stream>>>(Q, K, Wpq, ws);
  const int FILT[4] = {1, 3, 7, 9};
  for (int i = 0; i < 4; ++i) {
    const float* wq = (const float*)d_in[9 + 4 * i];
    const float* bq = (const float*)d_in[10 + 4 * i];
    const float* wk = (const float*)d_in[11 + 4 * i];
    const float* bk = (const float*)d_in[12 + 4 * i];
    k_conv<<<8, 256, 0, stream>>>(ws + OFF_PROJ,         wq, bq, ws + OFF_CONV + i * 65536,       FILT[i]);
    k_conv<<<8, 256, 0, stream>>>(ws + OFF_PROJ + 65536, wk, bk, ws + OFF_CONV + (4 + i) * 65536, FILT[i]);
  }
  k_bnstats<<<64, 256, 0, stream>>>(ws, bn_g, bn_b);
  k_cscale<<<1, 64, 0, stream>>>(Wbq, Wbk, ws);
  k_sortQ<<<64, 256, 0, stream>>>(ws);
  k_sortK<<<64, 256, 0, stream>>>(ws);
  k_attn<<<512, 256, 0, stream>>>(ws, V, out, out + 4194304);
}